// GCNLayer_52192442581980
// MI455X (gfx1250) — compile-verified
//
#include <hip/hip_runtime.h>

typedef __attribute__((ext_vector_type(2))) float v2f;
typedef __attribute__((ext_vector_type(8))) float v8f;

#define CH 128   // IN_CH == OUT_CH == 128

// ---------------------------------------------------------------------------
// Kernel 1: h[N,128] = x[N,128] @ W[128,128] + b   (full f32 via WMMA 16x16x4)
// 256 threads = 8 waves; each wave computes one 16-row x 128-col strip.
// W is staged in LDS (64 KB) in K-pair-interleaved layout: Wt[k/2][n][2] so a
// B fragment {W[k][n], W[k+1][n]} is one aligned ds_load_b64.
// ---------------------------------------------------------------------------
__global__ __launch_bounds__(256) void gcn_gemm_bias(
    const float* __restrict__ x, const float* __restrict__ W,
    const float* __restrict__ b, float* __restrict__ h,
    int n_nodes, int m_tiles)
{
    __shared__ float Wt[CH * CH];   // 64 KB, interleaved pairs

    // Cooperative interleave: 2048 quads (64 k-pairs x 32 n-quads), 8/thread.
    {
        int t = threadIdx.x;
        #pragma unroll
        for (int i = 0; i < 8; ++i) {
            int q  = t + i * 256;        // 0..2047
            int k2 = q >> 5;             // k-pair index 0..63
            int n0 = (q & 31) * 4;       // n quad base
            float4 r0 = *(const float4*)&W[(2 * k2    ) * CH + n0];
            float4 r1 = *(const float4*)&W[(2 * k2 + 1) * CH + n0];
            float4 s0 = {r0.x, r1.x, r0.y, r1.y};
            float4 s1 = {r0.z, r1.z, r0.w, r1.w};
            float4* dst = (float4*)&Wt[(k2 * CH + n0) * 2];
            dst[0] = s0;
            dst[1] = s1;
        }
    }
    __syncthreads();

    const int wave = threadIdx.x >> 5;
    const int lane = threadIdx.x & 31;
    const int mt   = blockIdx.x * 8 + wave;
    if (mt >= m_tiles) return;            // wave-uniform: EXEC stays all-ones

    const int m0    = mt * 16;
    const int ncol  = lane & 15;          // N index for B/C/D frags, M row for A
    const int hi    = lane >> 4;          // half-wave select
    const int khalf = hi * 2;             // A/B: lanes 16-31 hold K+2, K+3

    // A-fragment source row (clamped for tail safety; stores are guarded)
    int xrow = m0 + ncol;
    if (xrow >= n_nodes) xrow = n_nodes - 1;
    const float* xr = x + (size_t)xrow * CH;

    // Accumulators initialized with bias: C[m][n] = b[n], n = nt*16 + ncol
    v8f acc[8];
    #pragma unroll
    for (int nt = 0; nt < 8; ++nt) {
        float bv = b[nt * 16 + ncol];
        v8f t = {};
        t += bv;                          // scalar splat across all 8 elems
        acc[nt] = t;
    }

    // K loop: 32 steps of 4; one A-frag feeds 8 WMMAs (8 N-tiles).
    for (int k0 = 0; k0 < CH; k0 += 4) {
        v2f a = *(const v2f*)(xr + k0 + khalf);   // K=k0+khalf, k0+khalf+1
        // B base for this wave's K rows: pair index (k0+khalf)/2, col ncol
        const float* Wk = &Wt[((k0 + khalf) >> 1) * (CH * 2) + ncol * 2];
        #pragma unroll
        for (int nt = 0; nt < 8; ++nt) {
            v2f bb = *(const v2f*)(Wk + nt * 32);   // one ds_load_b64
            acc[nt] = __builtin_amdgcn_wmma_f32_16x16x4_f32(
                /*neg_a=*/false, a, /*neg_b=*/false, bb,
                /*c_mod=*/(short)0, acc[nt],
                /*reuse_a=*/false, /*reuse_b=*/false);
        }
    }

    // Store the 16x128 strip. C layout: VGPR v -> M = v + hi*8, N = ncol.
    #pragma unroll
    for (int v = 0; v < 8; ++v) {
        int mrow = m0 + v + hi * 8;
        if (mrow < n_nodes) {
            float* hr = h + (size_t)mrow * CH;
            #pragma unroll
            for (int nt = 0; nt < 8; ++nt)
                hr[nt * 16 + ncol] = acc[nt][v];
        }
    }
}

// ---------------------------------------------------------------------------
// Kernel 2: zero the output (harness poisons d_out with 0xAA).
// ---------------------------------------------------------------------------
__global__ void gcn_zero(float4* __restrict__ out, long long n4)
{
    long long i      = (long long)blockIdx.x * blockDim.x + threadIdx.x;
    long long stride = (long long)gridDim.x * blockDim.x;
    float4 z = {0.f, 0.f, 0.f, 0.f};
    for (; i < n4; i += stride) out[i] = z;
}

// ---------------------------------------------------------------------------
// Kernel 3: scatter-add. One wave per edge; each lane handles 4 channels.
// out[row] += h[col] * w  via global_atomic_add_f32 (resolves in 192 MB L2).
// ---------------------------------------------------------------------------
__global__ __launch_bounds__(256) void gcn_scatter(
    const float* __restrict__ h, const int* __restrict__ eidx,
    const float* __restrict__ ew, float* __restrict__ out, int n_edges)
{
    int gid  = blockIdx.x * 256 + threadIdx.x;
    int e    = gid >> 5;
    int lane = gid & 31;
    if (e >= n_edges) return;

    int   row = eidx[e];                       // edge_index[0][e]
    int   col = eidx[(size_t)n_edges + e];     // edge_index[1][e]
    float w   = ew[e];

    const float4 v = *(const float4*)(h + (size_t)col * CH + lane * 4);
    float* dst = out + (size_t)row * CH + lane * 4;
    unsafeAtomicAdd(dst + 0, v.x * w);
    unsafeAtomicAdd(dst + 1, v.y * w);
    unsafeAtomicAdd(dst + 2, v.z * w);
    unsafeAtomicAdd(dst + 3, v.w * w);
}

// ---------------------------------------------------------------------------
extern "C" void kernel_launch(void* const* d_in, const int* in_sizes, int n_in,
                              void* d_out, int out_size, void* d_ws, size_t ws_size,
                              hipStream_t stream)
{
    const float* x    = (const float*)d_in[0];   // [N,128]
    const int*   eidx = (const int*)  d_in[1];   // [2,E]
    const float* ew   = (const float*)d_in[2];   // [E]
    const float* W    = (const float*)d_in[3];   // [128,128]
    const float* b    = (const float*)d_in[4];   // [128]
    float* out = (float*)d_out;                  // [N,128]
    float* h   = (float*)d_ws;                   // [N,128] scratch (51.2 MB)

    const int n_nodes = in_sizes[0] / CH;
    const int n_edges = in_sizes[2];

    // 1) h = x @ W + b
    const int m_tiles = (n_nodes + 15) / 16;
    const int gblocks = (m_tiles + 7) / 8;
    gcn_gemm_bias<<<gblocks, 256, 0, stream>>>(x, W, b, h, n_nodes, m_tiles);

    // 2) out = 0
    long long n4 = (long long)out_size / 4;
    gcn_zero<<<2048, 256, 0, stream>>>((float4*)out, n4);

    // 3) scatter-add over edges (8 edges per 256-thread block)
    const int sblocks = (n_edges + 7) / 8;
    gcn_scatter<<<sblocks, 256, 0, stream>>>(h, eidx, ew, out, n_edges);
}